// NovaLinkPredictor_50190987821479
// MI455X (gfx1250) — compile-verified
//
#include <hip/hip_runtime.h>

// Problem constants (fixed by the reference harness)
#define H   128
#define FD  512
#define NUSERS 200000

typedef __attribute__((ext_vector_type(2))) float v2f;
typedef __attribute__((ext_vector_type(8))) float v8f;

// D(16x16,f32) = A(16x4,f32) x B(4x16,f32) + C   -- full-precision CDNA5 WMMA
__device__ __forceinline__ v8f wmma_f32_k4(v2f a, v2f b, v8f c) {
#if defined(__AMDGCN__) && __has_builtin(__builtin_amdgcn_wmma_f32_16x16x4_f32)
  return __builtin_amdgcn_wmma_f32_16x16x4_f32(false, a, false, b, (short)0, c,
                                               false, false);
#else
  // host-pass / non-gfx1250 placeholder (never executed on gfx1250)
  c[0] += a[0] * b[0];
  return c;
#endif
}

// Stage W[k0+kk, n] (kk,n in 0..127) transposed into LDS with a per-column
// rotation: lds[n*128 + ((kk + 4n) & 127)].  Exactly 64 KB; the rotation makes
// the hot-loop ds_load_b64 reads bank-conflict-free (banks (k+4j) mod 64) while
// keeping 8-byte pair alignment (rotation offset is even).
__device__ __forceinline__ void stage_wT(float* lds, const float* __restrict__ W,
                                         int k0) {
  const int t = threadIdx.x;
#pragma unroll
  for (int i = 0; i < (H * H) / 256; ++i) {      // 64 iterations, coalesced reads
    int idx = i * 256 + t;
    int kk  = idx >> 7;                          // 0..127
    int n   = idx & (H - 1);                     // 0..127
    lds[n * H + ((kk + 4 * n) & (H - 1))] = W[(size_t)(k0 + kk) * H + n];
  }
}

// Per-wave: acc[c] (c=0..7 column tiles of 16) += A_row_tile x W_chunk.
// One v2f A load feeds 8 WMMAs; B comes from LDS as conflict-free ds_load_b64.
__device__ __forceinline__ void gemm_chunk(v8f* acc, const float* __restrict__ Arow,
                                           const float* lds, int m, int khalf,
                                           float scale) {
  int nbasew[8], nrot[8];
#pragma unroll
  for (int c = 0; c < 8; ++c) {
    int n = c * 16 + m;
    nbasew[c] = n * H;
    nrot[c]   = (4 * n) & (H - 1);
  }
#pragma unroll 2
  for (int k = 0; k < H; k += 4) {
    v2f a = *(const v2f*)(Arow + k + khalf);
    a.x *= scale;
    a.y *= scale;
#pragma unroll
    for (int c = 0; c < 8; ++c) {
      v2f b = *(const v2f*)(lds + nbasew[c] + ((k + khalf + nrot[c]) & (H - 1)));
      acc[c] = wmma_f32_k4(a, b, acc[c]);
    }
  }
}

// out[n] = sum_k v[k] * W[k*H + n] (+ bias[n]) ; launch <<<1, H>>>
__global__ void vecmat_kernel(const float* __restrict__ v,
                              const float* __restrict__ W,
                              const float* __restrict__ bias,
                              float* __restrict__ out) {
  int n = threadIdx.x;
  float s = bias ? bias[n] : 0.0f;
  for (int k = 0; k < H; ++k) s += v[k] * W[k * H + n];
  out[n] = s;
}

// per-edge degree counts (float so downstream can divide)
__global__ void deg_kernel(const int* __restrict__ src, const int* __restrict__ dst,
                           float* __restrict__ cnt_u, float* __restrict__ cnt_m,
                           int nE) {
  int e = blockIdx.x * blockDim.x + threadIdx.x;
  if (e < nE) {
    atomicAdd(&cnt_u[src[e]], 1.0f);
    atomicAdd(&cnt_m[dst[e]], 1.0f);
  }
}

// Agg[sidx[e], :] += Xsrc[gidx[e], :]   (128 threads per edge, 2 edges / block)
__global__ void scatter_kernel(const float* __restrict__ Xsrc,
                               const int* __restrict__ gidx,
                               const int* __restrict__ sidx,
                               float* __restrict__ Agg, int nE) {
  int e = blockIdx.x * 2 + (threadIdx.x >> 7);
  int f = threadIdx.x & (H - 1);
  if (e < nE) {
    int g = gidx[e];
    int s = sidx[e];
    atomicAdd(&Agg[(size_t)s * H + f], Xsrc[(size_t)g * H + f]);
  }
}

// movie_x = movie_feats @ Wm + bm   (K = FD = 512, staged in 4 chunks of 128)
// Block: 8 waves x 16 rows = 128 rows, each wave covers all 128 output columns.
__global__ void __launch_bounds__(256)
movie_lin_kernel(const float* __restrict__ A, const float* __restrict__ W,
                 const float* __restrict__ bias, float* __restrict__ Out, int N) {
  __shared__ float wt[H * H];                    // 64 KB
  const int lane  = threadIdx.x & 31;
  const int wave  = threadIdx.x >> 5;
  const int m     = lane & 15;
  const int khalf = (lane >> 4) * 2;
  const int rowbase = blockIdx.x * 128 + wave * 16;
  const bool valid  = rowbase < N;               // whole 16-row tiles (N % 16 == 0)
  const int arow    = valid ? rowbase + m : 0;

  v8f acc[8];
#pragma unroll
  for (int c = 0; c < 8; ++c) acc[c] = (v8f){};

  for (int k0 = 0; k0 < FD; k0 += H) {
    __syncthreads();
    stage_wT(wt, W, k0);
    __syncthreads();
    if (valid) gemm_chunk(acc, A + (size_t)arow * FD + k0, wt, m, khalf, 1.0f);
  }
  if (!valid) return;

#pragma unroll
  for (int c = 0; c < 8; ++c) {
    const float bv = bias[c * 16 + m];
#pragma unroll
    for (int r = 0; r < 8; ++r) {
      int orow = rowbase + r + (lane >> 4) * 8;  // C/D layout: vgpr r, lane-half
      Out[(size_t)orow * H + c * 16 + m] = acc[c][r] + bv;
    }
  }
}

// Out = [Agg/max(cntA,1)] @ Wl + X @ Wr + bias + (maskcnt[row]>0 ? maskvec : 0)
// any of Agg / X / maskvec may be null; optional ReLU.  1/deg folded into A loads.
__global__ void __launch_bounds__(256)
sage_combine_kernel(const float* __restrict__ Agg, const float* __restrict__ cntA,
                    const float* __restrict__ Wl, const float* __restrict__ X,
                    const float* __restrict__ Wr, const float* __restrict__ bias,
                    const float* __restrict__ maskvec,
                    const float* __restrict__ maskcnt,
                    float* __restrict__ Out, int N, int do_relu) {
  __shared__ float wt[H * H];                    // 64 KB, reused Wl then Wr
  const int lane  = threadIdx.x & 31;
  const int wave  = threadIdx.x >> 5;
  const int m     = lane & 15;
  const int khalf = (lane >> 4) * 2;
  const int rowbase = blockIdx.x * 128 + wave * 16;
  const bool valid  = rowbase < N;
  const int arow    = valid ? rowbase + m : 0;

  v8f acc[8];
#pragma unroll
  for (int c = 0; c < 8; ++c) acc[c] = (v8f){};

  if (Agg) {                                     // uniform branch (kernel arg)
    stage_wT(wt, Wl, 0);
    __syncthreads();
    if (valid) {
      float scale = cntA ? 1.0f / fmaxf(cntA[arow], 1.0f) : 1.0f;
      gemm_chunk(acc, Agg + (size_t)arow * H, wt, m, khalf, scale);
    }
    __syncthreads();
  }
  if (X) {
    stage_wT(wt, Wr, 0);
    __syncthreads();
    if (valid) gemm_chunk(acc, X + (size_t)arow * H, wt, m, khalf, 1.0f);
  }
  if (!valid) return;

  float mflag[8];
#pragma unroll
  for (int r = 0; r < 8; ++r) {
    int orow = rowbase + r + (lane >> 4) * 8;
    mflag[r] = (maskvec && maskcnt[orow] > 0.0f) ? 1.0f : 0.0f;
  }
#pragma unroll
  for (int c = 0; c < 8; ++c) {
    const float bv = bias ? bias[c * 16 + m] : 0.0f;
    const float mv = maskvec ? maskvec[c * 16 + m] : 0.0f;
#pragma unroll
    for (int r = 0; r < 8; ++r) {
      int orow = rowbase + r + (lane >> 4) * 8;
      float v = acc[c][r] + bv + mv * mflag[r];
      if (do_relu) v = fmaxf(v, 0.0f);
      Out[(size_t)orow * H + c * 16 + m] = v;
    }
  }
}

// out[e] = dot(U[lu[e],:], M[lm[e],:])  over H=128 ; one wave32 per edge
__global__ void edge_dot_kernel(const float* __restrict__ U,
                                const float* __restrict__ M,
                                const int* __restrict__ lu,
                                const int* __restrict__ lm,
                                float* __restrict__ out, int nE) {
  int w = (blockIdx.x * blockDim.x + threadIdx.x) >> 5;
  int lane = threadIdx.x & 31;
  if (w >= nE) return;
  const float4* up = (const float4*)(U + (size_t)lu[w] * H);
  const float4* mp = (const float4*)(M + (size_t)lm[w] * H);
  float4 a = up[lane];
  float4 b = mp[lane];
  float s = a.x * b.x + a.y * b.y + a.z * b.z + a.w * b.w;
#pragma unroll
  for (int off = 16; off > 0; off >>= 1) s += __shfl_xor(s, off, 32);
  if (lane == 0) out[w] = s;
}

extern "C" void kernel_launch(void* const* d_in, const int* in_sizes, int n_in,
                              void* d_out, int out_size, void* d_ws, size_t ws_size,
                              hipStream_t stream) {
  (void)n_in; (void)out_size; (void)ws_size;
  const float* movie_feats = (const float*)d_in[0];
  const float* user_init   = (const float*)d_in[1];
  const int*   edge_src    = (const int*)d_in[2];
  const int*   edge_dst    = (const int*)d_in[3];
  const int*   lbl_user    = (const int*)d_in[4];
  const int*   lbl_movie   = (const int*)d_in[5];
  const float* Wm     = (const float*)d_in[7];
  const float* bm     = (const float*)d_in[8];
  const float* Wl1_um = (const float*)d_in[9];
  const float* bl1_um = (const float*)d_in[10];
  const float* Wr1_um = (const float*)d_in[11];
  const float* Wl1_mu = (const float*)d_in[12];
  const float* bl1_mu = (const float*)d_in[13];
  const float* Wr1_mu = (const float*)d_in[14];
  const float* Wl2_um = (const float*)d_in[15];
  const float* bl2_um = (const float*)d_in[16];
  const float* Wr2_um = (const float*)d_in[17];
  const float* Wl2_mu = (const float*)d_in[18];
  const float* bl2_mu = (const float*)d_in[19];
  const float* Wr2_mu = (const float*)d_in[20];

  const int E  = in_sizes[2];
  const int EL = in_sizes[4];
  const int NM = in_sizes[0] / FD;
  const int NU = NUSERS;

  // workspace carve-out (f32)
  float* ws = (float*)d_ws;
  size_t o = 0;
  float* movie_x  = ws + o; o += (size_t)NM * H;   // later reused as movie_agg2
  float* movie_h  = ws + o; o += (size_t)NM * H;
  float* movie_o  = ws + o; o += (size_t)NM * H;
  float* user_agg = ws + o; o += (size_t)NU * H;   // conv1 agg, then conv2 agg
  float* user_h   = ws + o; o += (size_t)NU * H;
  float* user_o   = ws + o; o += (size_t)NU * H;
  float* cnt_u    = ws + o; o += NU;
  float* cnt_m    = ws + o; o += NM;
  float* vec_u1   = ws + o; o += H;  // bl1_mu + user_init @ Wr1_mu (root collapses)
  float* vec_m1   = ws + o; o += H;  // user_init @ Wl1_um (agg collapses to mask)

  // zero accumulators (capture-safe memset nodes)
  hipMemsetAsync(cnt_u, 0, (size_t)NU * sizeof(float), stream);
  hipMemsetAsync(cnt_m, 0, (size_t)NM * sizeof(float), stream);
  hipMemsetAsync(user_agg, 0, (size_t)NU * H * sizeof(float), stream);

  vecmat_kernel<<<1, H, 0, stream>>>(user_init, Wr1_mu, bl1_mu, vec_u1);
  vecmat_kernel<<<1, H, 0, stream>>>(user_init, Wl1_um, nullptr, vec_m1);
  deg_kernel<<<(E + 255) / 256, 256, 0, stream>>>(edge_src, edge_dst, cnt_u, cnt_m, E);

  // movie_x = movie_feats @ Wm + bm
  movie_lin_kernel<<<(NM + 127) / 128, 256, 0, stream>>>(movie_feats, Wm, bm,
                                                         movie_x, NM);

  // conv1: user_agg[u] = sum over edges of movie_x[edge_dst] (scatter to edge_src)
  scatter_kernel<<<(E + 1) / 2, 256, 0, stream>>>(movie_x, edge_dst, edge_src,
                                                  user_agg, E);
  // user_h = relu( mean(user_agg) @ Wl1_mu + (bl1_mu + user_init@Wr1_mu) )
  sage_combine_kernel<<<(NU + 127) / 128, 256, 0, stream>>>(
      user_agg, cnt_u, Wl1_mu, nullptr, nullptr, vec_u1, nullptr, nullptr,
      user_h, NU, 1);
  // movie_h = relu( [indeg>0]*(user_init@Wl1_um) + bl1_um + movie_x @ Wr1_um )
  sage_combine_kernel<<<(NM + 127) / 128, 256, 0, stream>>>(
      nullptr, nullptr, nullptr, movie_x, Wr1_um, bl1_um, vec_m1, cnt_m,
      movie_h, NM, 1);

  // conv2 aggregations (reuse buffers after consumers finished; stream-ordered)
  hipMemsetAsync(movie_x, 0, (size_t)NM * H * sizeof(float), stream);   // movie_agg2
  hipMemsetAsync(user_agg, 0, (size_t)NU * H * sizeof(float), stream);  // user_agg2
  scatter_kernel<<<(E + 1) / 2, 256, 0, stream>>>(user_h, edge_src, edge_dst,
                                                  movie_x, E);
  scatter_kernel<<<(E + 1) / 2, 256, 0, stream>>>(movie_h, edge_dst, edge_src,
                                                  user_agg, E);

  // movie_o = mean(movie_agg2) @ Wl2_um + bl2_um + movie_h @ Wr2_um
  sage_combine_kernel<<<(NM + 127) / 128, 256, 0, stream>>>(
      movie_x, cnt_m, Wl2_um, movie_h, Wr2_um, bl2_um, nullptr, nullptr,
      movie_o, NM, 0);
  // user_o = mean(user_agg2) @ Wl2_mu + bl2_mu + user_h @ Wr2_mu
  sage_combine_kernel<<<(NU + 127) / 128, 256, 0, stream>>>(
      user_agg, cnt_u, Wl2_mu, user_h, Wr2_mu, bl2_mu, nullptr, nullptr,
      user_o, NU, 0);

  // out[e] = dot(user_o[lbl_user[e]], movie_o[lbl_movie[e]])
  edge_dot_kernel<<<(EL + 7) / 8, 256, 0, stream>>>(
      user_o, movie_o, lbl_user, lbl_movie, (float*)d_out, EL);
}